// dual_plane_43344809952172
// MI455X (gfx1250) — compile-verified
//
#include <hip/hip_runtime.h>

// dual-plane multires feature interp, MI455X (gfx1250).
// Bandwidth-bound: ~400MB HBM streaming; codebooks (~3MB) are L2/LDS-resident.
// LOD0 codebook is TDM-DMA'd into LDS *with pad insertion* (row stride 96->108
// dwords) so random-row ds_load_b128 gathers spread over all 16 bank groups.
// Output is staged per-wave in LDS and written back as perfectly coalesced
// non-temporal b128 stores (contiguous 512B per store instruction).

#define FEAT 48

typedef float    f4  __attribute__((ext_vector_type(4)));
typedef float    f2  __attribute__((ext_vector_type(2)));
typedef unsigned v4u __attribute__((ext_vector_type(4)));
typedef int      v8i __attribute__((ext_vector_type(8)));
typedef int      v4i __attribute__((ext_vector_type(4)));

struct BilinW {
  int x1, x2, y1, y2;
  float wx1, wx2, wy1, wy2;
};

__device__ __forceinline__ BilinW make_w(float px, float py, int res) {
  float fres = (float)(res - 1);
  float x = px * fres;
  float y = py * fres;
  float hi = fres - 1e-5f;
  float cx = fminf(fmaxf(x, 0.0f), hi);
  float cy = fminf(fmaxf(y, 0.0f), hi);
  int x1 = (int)cx;                  // cx >= 0 -> trunc == floor
  int y1 = (int)cy;
  int x2 = x1 + 1; if (x2 > res - 1) x2 = res - 1;
  int y2 = y1 + 1; if (y2 > res - 1) y2 = res - 1;
  BilinW w;
  w.x1 = x1; w.x2 = x2; w.y1 = y1; w.y2 = y2;
  w.wx1 = (float)x2 - x;  w.wx2 = x - (float)x1;
  w.wy1 = (float)y2 - y;  w.wy2 = y - (float)y1;
  return w;
}

// Global-memory LODs: rows are 96 floats (cb_x | cb_y), L2-hot.
__device__ __forceinline__ void accum_lod(f4 acc[12], const float* __restrict__ cb,
                                          const BilinW& w) {
  const float* rx1 = cb + (size_t)w.x1 * 96;
  const float* rx2 = cb + (size_t)w.x2 * 96;
  const float* ry1 = cb + (size_t)w.y1 * 96 + FEAT;
  const float* ry2 = cb + (size_t)w.y2 * 96 + FEAT;
#pragma unroll
  for (int c = 0; c < 12; ++c) {
    f4 a = *(const f4*)(rx1 + c * 4);
    f4 b = *(const f4*)(rx2 + c * 4);
    f4 e = *(const f4*)(ry1 + c * 4);
    f4 d = *(const f4*)(ry2 + c * 4);
    acc[c] += w.wx1 * a + w.wx2 * b + w.wy1 * e + w.wy2 * d;
  }
}

// LDS LOD0: TDM inserted a 4-dword pad after every 32 data dwords, so a 96-dword
// row occupies 108 dwords (row granule stride 27, coprime with 16 bank groups).
// Element at row r, dword d lives at r*108 + d + 4*(d/32).
__device__ __forceinline__ void accum_lod0(f4 acc[12], const float* lds,
                                           const BilinW& w) {
  int bx1 = w.x1 * 108, bx2 = w.x2 * 108;
  int by1 = w.y1 * 108, by2 = w.y2 * 108;
#pragma unroll
  for (int c = 0; c < 12; ++c) {
    const int dx = 4 * c + ((c >= 8) ? 4 : 0);            // d = 4c
    const int dy = 48 + 4 * c + ((c >= 4) ? 8 : 4);       // d = 48 + 4c
    f4 a = *(const f4*)(lds + bx1 + dx);
    f4 b = *(const f4*)(lds + bx2 + dx);
    f4 e = *(const f4*)(lds + by1 + dy);
    f4 d = *(const f4*)(lds + by2 + dy);
    acc[c] += w.wx1 * a + w.wx2 * b + w.wy1 * e + w.wy2 * d;
  }
}

__global__ void __launch_bounds__(512, 1)
dual_plane_43344809952172_kernel(const f2* __restrict__ pts,
                                 const float* __restrict__ cb0,
                                 const float* __restrict__ cb1,
                                 const float* __restrict__ cb2,
                                 const float* __restrict__ cb3,
                                 float* __restrict__ out, int n) {
  // Padded LOD0 codebook: 512 rows * 108 dwords = 216KB (+ tail pad slack).
  __shared__ __align__(16) float lds_cb0[55296];
  // Per-wave output staging: 16 waves * 32 pts * 12 f4 = 96KB. Total 312KB.
  __shared__ __align__(16) f4 stage4[512 * 12];

  // Wave 0 issues one TDM: 49152 f32 elements global->LDS with pad insertion
  // (pad_enable, interval=32 dwords, amount=4 dwords). TENSORcnt tracks it.
  if (threadIdx.x < 32) {
    unsigned long long ga = (unsigned long long)cb0;
    unsigned lds_base = (unsigned)(unsigned long long)&lds_cb0[0];
    v4u g0;
    g0.x = 1u;                                   // count=1 (valid), user mode
    g0.y = lds_base;                             // lds_addr (bytes)
    g0.z = (unsigned)(ga & 0xFFFFFFFFull);       // global_addr[31:0]
    g0.w = (unsigned)((ga >> 32) & 0x1FFFFFFull) // global_addr[56:32]
           | (2u << 30);                         // type = 2 ("image")
    v8i g1;
    g1[0] = 0x07120000;        // data_size=4B | pad_en | interval=32dw | pad=4dw
    g1[1] = (int)0xC0000000;   // tensor_dim0[15:0] = 49152 (<<16)
    g1[2] = 0x00010000;        // tensor_dim0 hi = 0, tensor_dim1 = 1 (<<16)
    g1[3] = (int)0xC0000000;   // tensor_dim1 hi = 0, tile_dim0 = 49152 (<<16)
    g1[4] = 1;                 // tile_dim1 = 1, tile_dim2 = 0
    g1[5] = 49152;             // tensor_dim0_stride[31:0]
    g1[6] = 0;                 // stride0 hi, stride1 lo
    g1[7] = 0;                 // stride1 hi
    v4i gz4 = {0, 0, 0, 0};
    v8i gz8 = {0, 0, 0, 0, 0, 0, 0, 0};
    __builtin_amdgcn_tensor_load_to_lds(g0, g1, gz4, gz4, gz8, 0);
    __builtin_amdgcn_s_wait_tensorcnt(0);
  }
  __syncthreads();

  const int lane = threadIdx.x & 31;
  const int wv   = threadIdx.x >> 5;
  f4* st = stage4 + wv * 384;              // wave-private staging window

  int tid = blockIdx.x * blockDim.x + threadIdx.x;
  int stride = gridDim.x * blockDim.x;
  // n % 32 == 0 and wave lanes hold consecutive p, so each wave is uniformly
  // active or inactive per iteration (staging exchange is whole-wave safe).
  for (int p = tid; p < n; p += stride) {
    f2 pt = __builtin_nontemporal_load(&pts[p]);   // streaming input: NT load
    f4 acc[12];
#pragma unroll
    for (int c = 0; c < 12; ++c) acc[c] = (f4)0.0f;

    { BilinW w = make_w(pt.x, pt.y, 512);  accum_lod0(acc, lds_cb0, w); } // LDS
    { BilinW w = make_w(pt.x, pt.y, 1024); accum_lod(acc, cb1, w); }      // L2
    { BilinW w = make_w(pt.x, pt.y, 2048); accum_lod(acc, cb2, w); }
    { BilinW w = make_w(pt.x, pt.y, 4096); accum_lod(acc, cb3, w); }

    // Stage this wave's 32 output rows (point-major), then write back linear:
    // each NT b128 store instruction covers a contiguous 512B of the output.
#pragma unroll
    for (int c = 0; c < 12; ++c) st[lane * 12 + c] = acc[c];
    __builtin_amdgcn_wave_barrier();       // LDS is in-order within a wave

    int pBase = p - lane;                  // uniform across the wave
    f4* ob = (f4*)out + (size_t)pBase * 12;
#pragma unroll
    for (int c = 0; c < 12; ++c)
      __builtin_nontemporal_store(st[c * 32 + lane], ob + c * 32 + lane);
    __builtin_amdgcn_wave_barrier();       // keep next iter's writes behind reads
  }
}

extern "C" void kernel_launch(void* const* d_in, const int* in_sizes, int n_in,
                              void* d_out, int out_size, void* d_ws, size_t ws_size,
                              hipStream_t stream) {
  const f2*    pts = (const f2*)d_in[0];
  const float* cb0 = (const float*)d_in[1];
  const float* cb1 = (const float*)d_in[2];
  const float* cb2 = (const float*)d_in[3];
  const float* cb3 = (const float*)d_in[4];
  float* out = (float*)d_out;
  int n = in_sizes[0] / 2;   // 2,000,000 points

  // 512 blocks x 512 threads: 312KB LDS -> 1 block/WGP (4 waves/SIMD32),
  // grid-stride ~8 points per thread.
  dual_plane_43344809952172_kernel<<<dim3(512), dim3(512), 0, stream>>>(
      pts, cb0, cb1, cb2, cb3, out, n);
}